// GCN_4492535792517
// MI455X (gfx1250) — compile-verified
//
#include <hip/hip_runtime.h>

typedef float v2f __attribute__((ext_vector_type(2)));
typedef float v8f __attribute__((ext_vector_type(8)));

#define NN 100000
#define EE 1600000
#define HLD 152   // padded row stride for h / hp (K=150 -> 152)
#define XLD 28    // padded row stride for x (K=25 -> 28)

static __device__ __forceinline__ float leakyf(float v, float s) {
    return v > 0.f ? v : v * s;
}

// float atomic max via signed/unsigned int trick (works with -inf init)
static __device__ __forceinline__ void atomicMaxF(float* addr, float v) {
    if (v >= 0.f)
        atomicMax((int*)addr, __float_as_int(v));
    else
        atomicMin((unsigned int*)addr, __float_as_uint(v));
}

// ---------------------------------------------------------------------------
// Packing kernels: zero-padded operands so the WMMA loop needs no guards.
// ---------------------------------------------------------------------------
__global__ void pack_weight(const float* __restrict__ W, const float* __restrict__ b,
                            float* __restrict__ Wp, float* __restrict__ bp,
                            int K, int N, int Kp, int Np)
{
    int t = blockIdx.x * blockDim.x + threadIdx.x;
    int total = Kp * Np;
    if (t < total) {
        int k = t / Np, n = t - k * Np;
        Wp[t] = (k < K && n < N) ? W[k * N + n] : 0.f;
    } else if (t < total + Np) {
        int n = t - total;
        bp[n] = (n < N) ? b[n] : 0.f;
    }
}

__global__ void pack_x(const float* __restrict__ x, float* __restrict__ xp)
{
    int t = blockIdx.x * blockDim.x + threadIdx.x;
    if (t >= NN * XLD) return;
    int i = t / XLD, k = t - i * XLD;
    xp[t] = (k < 25) ? x[i * 25 + k] : 0.f;
}

// zero pad columns 150,151 of h and hp (row stride 152)
__global__ void pad_zero(float* __restrict__ h, float* __restrict__ hp)
{
    int t = blockIdx.x * blockDim.x + threadIdx.x;
    if (t >= NN * 2) return;
    int i = t >> 1, k = 150 + (t & 1);
    h[(long)i * HLD + k]  = 0.f;
    hp[(long)i * HLD + k] = 0.f;
}

// ---------------------------------------------------------------------------
// fp32 WMMA GEMM on padded operands (Kp, Np compile-time -> fully unrolled,
// all loads become base + immediate offset):
//   C[M, 0..N) = act(A[M, Kp(pad0)] @ Wp[Kp, Np(pad0)] + bp)
// One wave -> one 16x16 tile. Requires M%16==0, Kp%4==0, Np%16==0, lda even.
// ---------------------------------------------------------------------------
template <int Kp, int Np>
__global__ void wmma_gemm_f32(const float* __restrict__ A, int lda,
                              const float* __restrict__ Wp,
                              const float* __restrict__ bp,
                              float* __restrict__ C, int ldc,
                              int N, float slope)
{
    const int lane = threadIdx.x & 31;
    const int wv   = threadIdx.x >> 5;
    const int m0   = blockIdx.x * 16;
    const int n0   = (blockIdx.y * (blockDim.x >> 5) + wv) * 16;
    if (n0 >= Np) return;                 // wave-uniform: EXEC stays full for WMMA
    const int half = lane >> 4;           // K-pair selector per ISA f32 A/B layout
    const int l    = lane & 15;

    const float* pA = A  + (long)(m0 + l) * lda + half * 2;   // even idx -> 8B aligned
    const float* pB = Wp + (long)(half * 2) * Np + (n0 + l);

    v8f c = {0.f, 0.f, 0.f, 0.f, 0.f, 0.f, 0.f, 0.f};
#pragma unroll
    for (int k = 0; k < Kp; k += 4) {
        v2f a = *(const v2f*)(pA + k);    // constant offsets after full unroll
        v2f b;
        b.x = pB[k * Np];
        b.y = pB[k * Np + Np];
        c = __builtin_amdgcn_wmma_f32_16x16x4_f32(false, a, false, b,
                                                  (short)0, c, false, false);
    }

    const int  ocol  = n0 + l;
    const bool colOK = ocol < N;
    const float bias = bp[ocol];
#pragma unroll
    for (int r = 0; r < 8; ++r) {
        const int orow = m0 + half * 8 + r;     // C/D layout: VGPR r -> rows r, r+8
        if (colOK) {
            float v = c[r] + bias;
            C[(long)orow * ldc + ocol] = leakyf(v, slope);
        }
    }
}

// ---------------------------------------------------------------------------
// GAT edge-phase kernels
// ---------------------------------------------------------------------------
__global__ void gat_init(float* __restrict__ h, float* __restrict__ smax,
                         float* __restrict__ ssum, int off)
{
    int tid = blockIdx.x * blockDim.x + threadIdx.x;
    if (tid >= NN * 50) return;
    int i = tid / 50, k = tid - i * 50;
    h[(long)i * HLD + off + k] = 0.f;
    if (k == 0) { smax[i] = -INFINITY; ssum[i] = 0.f; }
}

// score_e = leaky(xl[src]+xr[dst], 0.2) . att ; smax[dst] = max(...)
__global__ void edge_score(const int* __restrict__ src, const int* __restrict__ dst,
                           const float* __restrict__ xl, const float* __restrict__ xr,
                           const float* __restrict__ att,
                           float* __restrict__ score, float* __restrict__ smax)
{
    int t = blockIdx.x * blockDim.x + threadIdx.x;
    const int total = EE + NN;
    if (t >= total) return;
    int s, d;
    if (t < EE) { s = src[t]; d = dst[t]; } else { s = d = t - EE; }
    const v2f* pl = (const v2f*)(xl + (long)s * 50);   // 50-float rows: 8B aligned
    const v2f* pr = (const v2f*)(xr + (long)d * 50);
    float acc = 0.f;
#pragma unroll
    for (int k = 0; k < 25; ++k) {
        v2f a = pl[k], b = pr[k];
        acc = fmaf(leakyf(a.x + b.x, 0.2f), att[2 * k],     acc);
        acc = fmaf(leakyf(a.y + b.y, 0.2f), att[2 * k + 1], acc);
    }
    score[t] = acc;
    atomicMaxF(&smax[d], acc);
}

// w_e = exp(score - smax[dst]); ssum[dst] += w ; score <- w
__global__ void edge_exp(const int* __restrict__ dst,
                         float* __restrict__ score,
                         const float* __restrict__ smax, float* __restrict__ ssum)
{
    int t = blockIdx.x * blockDim.x + threadIdx.x;
    const int total = EE + NN;
    if (t >= total) return;
    int d = (t < EE) ? dst[t] : t - EE;
    float w = __expf(score[t] - smax[d]);
    score[t] = w;
    atomicAdd(&ssum[d], w);
}

// h[dst, off+k] += w_e * xl[src, k]   (k-major: one atomic per thread, coalesced)
__global__ void edge_scatter(const int* __restrict__ src, const int* __restrict__ dst,
                             const float* __restrict__ w, const float* __restrict__ xl,
                             float* __restrict__ h, int off)
{
    long tid = (long)blockIdx.x * blockDim.x + threadIdx.x;
    const long total2 = (long)(EE + NN) * 50;
    if (tid >= total2) return;
    int e = (int)(tid / 50);
    int k = (int)(tid - (long)e * 50);
    int s, d;
    if (e < EE) { s = src[e]; d = dst[e]; } else { s = d = e - EE; }
    atomicAdd(&h[(long)d * HLD + off + k], w[e] * xl[(long)s * 50 + k]);
}

// h <- leaky( h / (ssum + 1e-16) + bo, 0.1 )
__global__ void gat_final(float* __restrict__ h, const float* __restrict__ ssum,
                          const float* __restrict__ bo, int off)
{
    int tid = blockIdx.x * blockDim.x + threadIdx.x;
    if (tid >= NN * 50) return;
    int i = tid / 50, k = tid - i * 50;
    long idx = (long)i * HLD + off + k;
    float v = h[idx] / (ssum[i] + 1e-16f) + bo[k];
    h[idx] = leakyf(v, 0.1f);
}

// ---------------------------------------------------------------------------
// Fused tail: out = act(h1 @ W2 + b2) @ W3 + b3   (75 -> 30 -> 2), per thread
// ---------------------------------------------------------------------------
__global__ void cls_tail(const float* __restrict__ h1,
                         const float* __restrict__ W2, const float* __restrict__ b2,
                         const float* __restrict__ W3, const float* __restrict__ b3,
                         float* __restrict__ out)
{
    __shared__ float sW2[75 * 30], sW3[60], sb2[30], sb3[2];
    for (int t = threadIdx.x; t < 75 * 30; t += blockDim.x) sW2[t] = W2[t];
    if (threadIdx.x < 60) sW3[threadIdx.x] = W3[threadIdx.x];
    if (threadIdx.x < 30) sb2[threadIdx.x] = b2[threadIdx.x];
    if (threadIdx.x < 2)  sb3[threadIdx.x] = b3[threadIdx.x];
    __syncthreads();

    int i = blockIdx.x * blockDim.x + threadIdx.x;
    if (i >= NN) return;
    float acc[30];
#pragma unroll
    for (int j = 0; j < 30; ++j) acc[j] = sb2[j];
    const float* row = h1 + (long)i * 75;
    for (int k = 0; k < 75; ++k) {
        float hk = row[k];
#pragma unroll
        for (int j = 0; j < 30; ++j) acc[j] = fmaf(hk, sW2[k * 30 + j], acc[j]);
    }
    float o0 = sb3[0], o1 = sb3[1];
#pragma unroll
    for (int j = 0; j < 30; ++j) {
        float a = leakyf(acc[j], 0.1f);
        o0 = fmaf(a, sW3[j * 2 + 0], o0);
        o1 = fmaf(a, sW3[j * 2 + 1], o1);
    }
    out[(long)i * 2 + 0] = o0;
    out[(long)i * 2 + 1] = o1;
}

// ---------------------------------------------------------------------------
extern "C" void kernel_launch(void* const* d_in, const int* in_sizes, int n_in,
                              void* d_out, int out_size, void* d_ws, size_t ws_size,
                              hipStream_t stream)
{
    const float* x = (const float*)d_in[0];
    const int* ei[3] = {(const int*)d_in[1], (const int*)d_in[2], (const int*)d_in[3]};

    // ---- workspace layout (floats) ----
    float* ws    = (float*)d_ws;
    float* h     = ws;                          // [NN, 152]
    float* hp    = h    + (long)NN * HLD;       // [NN, 152]
    float* xpad  = hp   + (long)NN * HLD;       // [NN, 28]
    float* xl    = xpad + (long)NN * XLD;       // [NN, 50]
    float* xr    = xl   + (long)NN * 50;        // [NN, 50]
    float* score = xr   + (long)NN * 50;        // [EE+NN]
    float* smax  = score + (EE + NN);           // [NN]
    float* ssum  = smax + NN;                   // [NN]
    float* wcur  = ssum + NN;                   // packed weights region
    float* h1    = xl;                          // [NN,75] alias after GAT phase

    // packed weight buffers
    float* Wlp[3]; float* blp[3]; float* Wrp[3]; float* brp[3];
    for (int r = 0; r < 3; ++r) {
        Wlp[r] = wcur; wcur += 28 * 64;  blp[r] = wcur; wcur += 64;
        Wrp[r] = wcur; wcur += 28 * 64;  brp[r] = wcur; wcur += 64;
    }
    float* projp = wcur; wcur += 152 * 160;  float* projb = wcur; wcur += 160;
    float* cls1p = wcur; wcur += 152 * 80;   float* cls1b = wcur; wcur += 80;

    const dim3 blkG(128);                       // 4 waves / block
    const int  total  = EE + NN;
    const long total2 = (long)total * 50;
    const int  gEdge  = (total + 255) / 256;
    const int  gEdge2 = (int)((total2 + 255) / 256);
    const int  gNode  = (NN * 50 + 255) / 256;

    // ---- pack all padded operands up front ----
    pack_x<<<(NN * XLD + 255) / 256, 256, 0, stream>>>(x, xpad);
    pad_zero<<<(NN * 2 + 255) / 256, 256, 0, stream>>>(h, hp);
    for (int r = 0; r < 3; ++r) {
        pack_weight<<<(28 * 64 + 64 + 255) / 256, 256, 0, stream>>>(
            (const float*)d_in[4 + 6 * r + 0], (const float*)d_in[4 + 6 * r + 1],
            Wlp[r], blp[r], 25, 50, 28, 64);
        pack_weight<<<(28 * 64 + 64 + 255) / 256, 256, 0, stream>>>(
            (const float*)d_in[4 + 6 * r + 2], (const float*)d_in[4 + 6 * r + 3],
            Wrp[r], brp[r], 25, 50, 28, 64);
    }
    pack_weight<<<(152 * 160 + 160 + 255) / 256, 256, 0, stream>>>(
        (const float*)d_in[22], (const float*)d_in[23], projp, projb, 150, 150, 152, 160);
    pack_weight<<<(152 * 80 + 80 + 255) / 256, 256, 0, stream>>>(
        (const float*)d_in[24], (const float*)d_in[25], cls1p, cls1b, 150, 75, 152, 80);

    // ---- three GATv2 relations ----
    for (int r = 0; r < 3; ++r) {
        const float* att = (const float*)d_in[4 + 6 * r + 4];
        const float* bo  = (const float*)d_in[4 + 6 * r + 5];
        const int* src = ei[r];
        const int* dst = ei[r] + EE;
        const int  off = 50 * r;

        gat_init<<<gNode, 256, 0, stream>>>(h, smax, ssum, off);
        // xl = x@Wl+bl, xr = x@Wr+br   (6250 row tiles; Np=64 -> 4 waves)
        wmma_gemm_f32<28, 64><<<dim3(6250, 1), blkG, 0, stream>>>(
            xpad, XLD, Wlp[r], blp[r], xl, 50, 50, 1.0f);
        wmma_gemm_f32<28, 64><<<dim3(6250, 1), blkG, 0, stream>>>(
            xpad, XLD, Wrp[r], brp[r], xr, 50, 50, 1.0f);
        edge_score<<<gEdge, 256, 0, stream>>>(src, dst, xl, xr, att, score, smax);
        edge_exp<<<gEdge, 256, 0, stream>>>(dst, score, smax, ssum);
        edge_scatter<<<gEdge2, 256, 0, stream>>>(src, dst, score, xl, h, off);
        gat_final<<<gNode, 256, 0, stream>>>(h, ssum, bo, off);
    }

    // hp = h @ proj_W + proj_b     (Np=160 -> 10 tiles -> grid.y=3)
    wmma_gemm_f32<152, 160><<<dim3(6250, 3), blkG, 0, stream>>>(
        h, HLD, projp, projb, hp, HLD, 150, 1.0f);
    // h1 = leaky(hp @ cls_W1 + cls_b1)  (Np=80 -> 5 tiles -> grid.y=2)
    wmma_gemm_f32<152, 80><<<dim3(6250, 2), blkG, 0, stream>>>(
        hp, HLD, cls1p, cls1b, h1, 75, 75, 0.1f);
    // out = leaky(h1@W2+b2)@W3+b3
    cls_tail<<<(NN + 255) / 256, 256, 0, stream>>>(
        h1, (const float*)d_in[26], (const float*)d_in[27],
        (const float*)d_in[28], (const float*)d_in[29], (float*)d_out);
}